// SGConv_5179730559344
// MI455X (gfx1250) — compile-verified
//
#include <hip/hip_runtime.h>

#define N_CH 128   // IN_CH == OUT_CH == 128

typedef __attribute__((ext_vector_type(2))) float v2f;
typedef __attribute__((ext_vector_type(8))) float v8f;

// ---------------------------------------------------------------------------
// Zero an int buffer (for the per-call degree histogram).
// ---------------------------------------------------------------------------
__global__ void zero_i32_kernel(int* __restrict__ p, int n) {
    int i = blockIdx.x * blockDim.x + threadIdx.x;
    if (i < n) p[i] = 0;
}

// ---------------------------------------------------------------------------
// Histogram of destination nodes: count[d] = in-degree of node d.
// ---------------------------------------------------------------------------
__global__ void hist_kernel(const int* __restrict__ dst,
                            int* __restrict__ count, int n_edges) {
    int e = blockIdx.x * blockDim.x + threadIdx.x;
    if (e < n_edges) atomicAdd(&count[dst[e]], 1);
}

// ---------------------------------------------------------------------------
// Single-workgroup exclusive scan: row_ptr[i] = sum_{k<i} count[k],
// row_ptr[n] = total. 1024 threads, Hillis-Steele in LDS, serial carry.
// n = 100000 -> 98 chunks; runs in microseconds.
// ---------------------------------------------------------------------------
__global__ void __launch_bounds__(1024)
scan_kernel(const int* __restrict__ count, int* __restrict__ row_ptr, int n) {
    __shared__ int sdata[1024];
    __shared__ int carry;
    const int tid = threadIdx.x;
    if (tid == 0) carry = 0;
    __syncthreads();
    for (int base = 0; base < n; base += 1024) {
        int i = base + tid;
        int v = (i < n) ? count[i] : 0;
        sdata[tid] = v;
        __syncthreads();
        for (int off = 1; off < 1024; off <<= 1) {
            int t = (tid >= off) ? sdata[tid - off] : 0;
            __syncthreads();
            sdata[tid] += t;
            __syncthreads();
        }
        int incl  = sdata[tid];      // inclusive scan of this chunk
        int total = sdata[1023];     // chunk sum
        if (i < n) row_ptr[i] = carry + incl - v;   // exclusive
        __syncthreads();             // all threads done reading carry
        if (tid == 0) carry += total;
        __syncthreads();             // carry update visible
    }
    if (tid == 0) row_ptr[n] = carry;
}

// ---------------------------------------------------------------------------
// cursor[i] = row_ptr[i]  (per-node write cursor for CSR fill).
// ---------------------------------------------------------------------------
__global__ void copy_i32_kernel(const int* __restrict__ src,
                                int* __restrict__ dstp, int n) {
    int i = blockIdx.x * blockDim.x + threadIdx.x;
    if (i < n) dstp[i] = src[i];
}

// ---------------------------------------------------------------------------
// Fill CSR: group edge source indices by destination node.
// ---------------------------------------------------------------------------
__global__ void fill_csr_kernel(const int* __restrict__ src,
                                const int* __restrict__ dst,
                                int* __restrict__ cursor,
                                int* __restrict__ csr_src, int n_edges) {
    int e = blockIdx.x * blockDim.x + threadIdx.x;
    if (e >= n_edges) return;
    int d = dst[e];
    int slot = atomicAdd(&cursor[d], 1);
    csr_src[slot] = src[e];
}

// ---------------------------------------------------------------------------
// Gather-mean (one hop, NO atomics):
//   out[n] = (1/max(deg,1)) * sum_{j in [row_ptr[n],row_ptr[n+1])} h[csr_src[j]]
// One wave32 per node; lane owns 4 consecutive channels (float4, fully
// coalesced 512B per gathered row). 2-way unrolled dual accumulators for ILP.
// ---------------------------------------------------------------------------
__global__ void gather_mean_kernel(const float* __restrict__ h,
                                   const int* __restrict__ csr_src,
                                   const int* __restrict__ row_ptr,
                                   float* __restrict__ out, int n_nodes) {
    long long gid = (long long)blockIdx.x * blockDim.x + threadIdx.x;
    int node = (int)(gid >> 5);
    if (node >= n_nodes) return;
    int lane = (int)(gid & 31);

    int beg = row_ptr[node];
    int end = row_ptr[node + 1];

    float a0 = 0.f, a1 = 0.f, a2 = 0.f, a3 = 0.f;  // accumulator 0
    float b0 = 0.f, b1 = 0.f, b2 = 0.f, b3 = 0.f;  // accumulator 1

    int j = beg;
    for (; j + 2 <= end; j += 2) {
        int s0 = csr_src[j];
        int s1 = csr_src[j + 1];
        const float4 v0 = *(const float4*)(h + (long long)s0 * N_CH + lane * 4);
        const float4 v1 = *(const float4*)(h + (long long)s1 * N_CH + lane * 4);
        a0 += v0.x; a1 += v0.y; a2 += v0.z; a3 += v0.w;
        b0 += v1.x; b1 += v1.y; b2 += v1.z; b3 += v1.w;
    }
    if (j < end) {
        int s = csr_src[j];
        const float4 v = *(const float4*)(h + (long long)s * N_CH + lane * 4);
        a0 += v.x; a1 += v.y; a2 += v.z; a3 += v.w;
    }

    float inv = 1.0f / fmaxf((float)(end - beg), 1.0f);
    float4 r;
    r.x = (a0 + b0) * inv;
    r.y = (a1 + b1) * inv;
    r.z = (a2 + b2) * inv;
    r.w = (a3 + b3) * inv;
    *(float4*)(out + (long long)node * N_CH + lane * 4) = r;
}

// ---------------------------------------------------------------------------
// out = H @ W^T + b using V_WMMA_F32_16X16X4_F32 (full f32 precision).
// Block = 256 threads = 8 waves. blockIdx.x = 16-node tile; wave w handles
// output-channel tile [16*w, 16*w+16). K-loop over 128 in steps of 4.
//
// A (16x4 f32): lane m = tid&15, VGPR pair holds K = 2*(tid>>4) + {0,1}.
// B (4x16 f32): lane n = tid&15, same K split (B(k,n) = W[n][k]).
// C/D (16x16 f32, 8 VGPRs): n = tid&15, m = r + 8*(tid>>4).
// ---------------------------------------------------------------------------
__global__ void __launch_bounds__(256)
gemm_wmma_kernel(const float* __restrict__ H, const float* __restrict__ W,
                 const float* __restrict__ bias, float* __restrict__ out,
                 int n_nodes) {
    const int tid   = threadIdx.x & 31;   // lane in wave
    const int wave  = threadIdx.x >> 5;   // 0..7 -> output-channel tile
    const int node0 = blockIdx.x * 16;
    const int o0    = wave * 16;

    const int mn = tid & 15;              // m for A, n for B
    const int kp = (tid >> 4) * 2;        // 0 or 2 (half-wave K split)

    const float* arow = H + (long long)(node0 + mn) * N_CH + kp;
    const float* brow = W + (long long)(o0 + mn) * N_CH + kp;   // B = W^T

    v8f acc = {};
#pragma unroll
    for (int k0 = 0; k0 < N_CH; k0 += 4) {
        v2f a, b;
        a.x = arow[k0];     a.y = arow[k0 + 1];
        b.x = brow[k0];     b.y = brow[k0 + 1];
        // 8 args: (neg_a, A, neg_b, B, c_mod, C, reuse_a, reuse_b)
        acc = __builtin_amdgcn_wmma_f32_16x16x4_f32(
            false, a, false, b, (short)0, acc, false, false);
    }

    const int n     = tid & 15;
    const int mbase = (tid >> 4) * 8;
    const float bv  = bias[o0 + n];
#pragma unroll
    for (int r = 0; r < 8; ++r) {
        out[(long long)(node0 + mbase + r) * N_CH + o0 + n] = acc[r] + bv;
    }
}

// ---------------------------------------------------------------------------
// Launch: build CSR (hist -> scan -> fill) once, then two atomic-free
// gather-mean hops, then the f32 WMMA GEMM.
// ws layout: accA [N*128 f32] | accB [N*128 f32] | row_ptr [(N+1) i32]
//            | tmp/cursor [N i32] | csr_src [E i32]
// ---------------------------------------------------------------------------
extern "C" void kernel_launch(void* const* d_in, const int* in_sizes, int n_in,
                              void* d_out, int out_size, void* d_ws, size_t ws_size,
                              hipStream_t stream) {
    const float* x   = (const float*)d_in[0];
    const int*   ei  = (const int*)d_in[1];   // [2, E] row-major
    const float* W   = (const float*)d_in[2]; // [128, 128]
    const float* b   = (const float*)d_in[3]; // [128]
    float*       out = (float*)d_out;

    const int n_nodes = in_sizes[0] / N_CH;
    const int n_edges = in_sizes[1] / 2;
    const int* src = ei;
    const int* dst = ei + n_edges;

    float* accA    = (float*)d_ws;
    float* accB    = accA + (long long)n_nodes * N_CH;
    int*   row_ptr = (int*)(accB + (long long)n_nodes * N_CH);
    int*   tmp     = row_ptr + (n_nodes + 1);   // count, then cursor
    int*   csr_src = tmp + n_nodes;

    const int BS = 256;
    const int ngrid_i = (n_nodes + BS - 1) / BS;          // per-node i32 ops
    const int egrid   = (n_edges + BS - 1) / BS;          // per-edge ops
    const int ggrid   = (int)(((long long)n_nodes * 32 + BS - 1) / BS); // wave/node
    const int mgrid   = n_nodes / 16;                     // 6250, exact

    // --- Build CSR (edges grouped by dst) -----------------------------------
    zero_i32_kernel<<<ngrid_i, BS, 0, stream>>>(tmp, n_nodes);
    hist_kernel<<<egrid, BS, 0, stream>>>(dst, tmp, n_edges);
    scan_kernel<<<1, 1024, 0, stream>>>(tmp, row_ptr, n_nodes);
    copy_i32_kernel<<<ngrid_i, BS, 0, stream>>>(row_ptr, tmp, n_nodes);
    fill_csr_kernel<<<egrid, BS, 0, stream>>>(src, dst, tmp, csr_src, n_edges);

    // --- Hop 1: x -> accA,  Hop 2: accA -> accB (no atomics) ----------------
    gather_mean_kernel<<<ggrid, BS, 0, stream>>>(x,    csr_src, row_ptr, accA, n_nodes);
    gather_mean_kernel<<<ggrid, BS, 0, stream>>>(accA, csr_src, row_ptr, accB, n_nodes);

    // --- out = accB @ W^T + b  (f32 WMMA) -----------------------------------
    gemm_wmma_kernel<<<mgrid, BS, 0, stream>>>(accB, W, b, out, n_nodes);
}